// GCNSpatial_53601191854857
// MI455X (gfx1250) — compile-verified
//
#include <hip/hip_runtime.h>

// CDNA5 / gfx1250: wave32. WMMA F32 16x16x4: A=16x4 f32 (2 VGPRs/lane),
// B=4x16 f32 (2 VGPRs/lane), C/D=16x16 f32 (8 VGPRs/lane).
typedef __attribute__((ext_vector_type(2))) float v2f;
typedef __attribute__((ext_vector_type(8))) float v8f;

static inline int cdiv_i(long long a, int b) { return (int)((a + b - 1) / b); }

__global__ void k_zero(float* __restrict__ p, int n) {
  int i = blockIdx.x * blockDim.x + threadIdx.x;
  if (i < n) p[i] = 0.0f;
}

// Degree at dst (self-loop added later as +1) and optional int64->int32 edge compression.
__global__ void k_deg(const long long* __restrict__ ei, float* __restrict__ deg,
                      int* __restrict__ s32, int* __restrict__ d32,
                      int e_count, int do_compress) {
  int e = blockIdx.x * blockDim.x + threadIdx.x;
  if (e >= e_count) return;
  int s = (int)ei[e];
  int d = (int)ei[(long long)e_count + e];
  atomicAdd(&deg[d], 1.0f);  // counts <= 16M: exact in f32
  if (do_compress) { s32[e] = s; d32[e] = d; }
}

__global__ void k_dinv(float* __restrict__ dinv, int n) {
  int v = blockIdx.x * blockDim.x + threadIdx.x;
  if (v < n) dinv[v] = rsqrtf(dinv[v] + 1.0f);  // +1 self-loop; always > 0
}

// Layer-1 linear is an outer product: g1[v] = dinv[v] * x[v] * W1[0:4]
__global__ void k_g1(const float* __restrict__ x, const float* __restrict__ W1,
                     const float* __restrict__ dinv, float4* __restrict__ g, int n) {
  int v = blockIdx.x * blockDim.x + threadIdx.x;
  if (v >= n) return;
  float s = dinv[v] * x[v];
  g[v] = make_float4(s * W1[0], s * W1[1], s * W1[2], s * W1[3]);
}

// Edge scatter: acc[dst] += g[src]. Node arrays are L2-resident (8 MB vs 192 MB L2);
// the edge stream is the HBM-bound part.
template <typename IdxT>
__global__ void k_scatter(const IdxT* __restrict__ src, const IdxT* __restrict__ dst,
                          const float4* __restrict__ g, float* __restrict__ acc, int e_count) {
  int e = blockIdx.x * blockDim.x + threadIdx.x;
  if (e >= e_count) return;
  int s = (int)src[e];
  int d = (int)dst[e];
  float4 m = g[s];
  float* a = acc + (size_t)d * 4;
  atomicAdd(a + 0, m.x);
  atomicAdd(a + 1, m.y);
  atomicAdd(a + 2, m.z);
  atomicAdd(a + 3, m.w);
}

// h'' = dinv * relu(dinv*(acc+g) + b1); also re-zero acc (reused for layer 2).
__global__ void k_finalize1(const float* __restrict__ dinv, const float4* __restrict__ g,
                            float4* __restrict__ acc, const float* __restrict__ b1,
                            float4* __restrict__ h, int n) {
  int v = blockIdx.x * blockDim.x + threadIdx.x;
  if (v >= n) return;
  float di = dinv[v];
  float4 a = acc[v];
  float4 gg = g[v];
  float4 r;
  r.x = di * fmaxf(di * (a.x + gg.x) + b1[0], 0.0f);
  r.y = di * fmaxf(di * (a.y + gg.y) + b1[1], 0.0f);
  r.z = di * fmaxf(di * (a.z + gg.z) + b1[2], 0.0f);
  r.w = di * fmaxf(di * (a.w + gg.w) + b1[3], 0.0f);
  h[v] = r;
  acc[v] = make_float4(0.0f, 0.0f, 0.0f, 0.0f);
}

// Layer-2 linear via V_WMMA_F32_16X16X4_F32: one wave transforms 16 nodes.
// A (16x4, MxK): lanes 0-15 hold row M=lane with K={0,1}; lanes 16-31 K={2,3}.
// B (4x16, KxN): W2 in columns N=0..3, zero elsewhere (mirrored layout).
// D (16x16 f32): VGPR r -> M=r (lanes 0-15) / M=r+8 (lanes 16-31), N=lane&15.
__global__ void k_linear2_wmma(const float* __restrict__ h, const float* __restrict__ W2,
                               float* __restrict__ g, int n) {
  const int lane = threadIdx.x & 31;
  const int tile = blockIdx.x * (blockDim.x >> 5) + (threadIdx.x >> 5);
  const int numTiles = (n + 15) >> 4;
  if (tile >= numTiles) return;  // wave-uniform: EXEC all-1s at the WMMA

  const int n0 = tile << 4;
  const int m = lane & 15;          // A row / B-D column index
  const int half = lane >> 4;       // 0 or 1
  const int kb = half << 1;         // K pair base: 0 or 2
  const bool full = (n0 + 16) <= n; // wave-uniform (always true when n % 16 == 0)

  // B: unconditional in-bounds load, then cndmask to zero (no exec-mask branches).
  // W2 is [Cin=4, Cout=4] row-major.
  float w0 = W2[kb * 4 + (m & 3)];
  float w1 = W2[(kb + 1) * 4 + (m & 3)];
  v2f b;
  b.x = (m < 4) ? w0 : 0.0f;
  b.y = (m < 4) ? w1 : 0.0f;

  // A: contiguous pair -> single global_load_b64 in the full-tile path.
  v2f a;
  if (full) {
    a = *(const v2f*)(h + (size_t)(n0 + m) * 4 + kb);
  } else {
    int node = n0 + m;
    int cl = (node < n) ? node : 0;  // clamped load + select, not exec branch
    v2f t = *(const v2f*)(h + (size_t)cl * 4 + kb);
    a.x = (node < n) ? t.x : 0.0f;
    a.y = (node < n) ? t.y : 0.0f;
  }

  v8f c = {};
  v8f d = __builtin_amdgcn_wmma_f32_16x16x4_f32(
      /*neg_a=*/false, a, /*neg_b=*/false, b,
      /*c_mod=*/(short)0, c, /*reuse_a=*/false, /*reuse_b=*/false);

  // One exec mask (m < 4), 8 plain stores in the full path.
  if (m < 4) {
    const int mbase = half << 3;  // rows 0..7 or 8..15
    float* gp = g + (size_t)(n0 + mbase) * 4 + m;
    if (full) {
#pragma unroll
      for (int r = 0; r < 8; ++r) gp[(size_t)r * 4] = d[r];
    } else {
#pragma unroll
      for (int r = 0; r < 8; ++r) {
        if (n0 + mbase + r < n) gp[(size_t)r * 4] = d[r];
      }
    }
  }
}

__global__ void k_finalize2(const float* __restrict__ dinv, const float4* __restrict__ g,
                            float4* __restrict__ out, const float* __restrict__ b2, int n) {
  int v = blockIdx.x * blockDim.x + threadIdx.x;
  if (v >= n) return;
  float di = dinv[v];
  float4 a = out[v];
  float4 gg = g[v];
  out[v] = make_float4(di * (a.x + gg.x) + b2[0],
                       di * (a.y + gg.y) + b2[1],
                       di * (a.z + gg.z) + b2[2],
                       di * (a.w + gg.w) + b2[3]);
}

extern "C" void kernel_launch(void* const* d_in, const int* in_sizes, int n_in,
                              void* d_out, int out_size, void* d_ws, size_t ws_size,
                              hipStream_t stream) {
  const float* x       = (const float*)d_in[0];
  const long long* ei  = (const long long*)d_in[1];  // [2, E] int64 (src row, dst row)
  const float* W1      = (const float*)d_in[2];
  const float* b1      = (const float*)d_in[3];
  const float* W2      = (const float*)d_in[4];
  const float* b2      = (const float*)d_in[5];
  float* out = (float*)d_out;

  const int n = in_sizes[0];       // x is [N,1]
  const int e = in_sizes[1] / 2;   // edge_index is [2,E]

  // Workspace layout: dinv [n] | g [4n] | h [4n] | (optional) src32 [e] dst32 [e]
  char* ws = (char*)d_ws;
  float*  dinv = (float*)ws;
  float4* g    = (float4*)(ws + (size_t)n * sizeof(float));
  float4* h    = (float4*)(ws + (size_t)n * 5 * sizeof(float));
  size_t base  = ((size_t)n * 9 * sizeof(float) + 255) & ~(size_t)255;
  const bool compress = ws_size >= base + (size_t)e * 2 * sizeof(int);
  int* s32 = compress ? (int*)(ws + base) : nullptr;
  int* d32 = compress ? s32 + e : nullptr;

  const int B = 256;

  // d_out doubles as the scatter accumulator for both layers (zeroed here and in finalize1).
  k_zero<<<cdiv_i(n, B), B, 0, stream>>>(dinv, n);
  k_zero<<<cdiv_i((long long)n * 4, B), B, 0, stream>>>(out, n * 4);

  k_deg<<<cdiv_i(e, B), B, 0, stream>>>(ei, dinv, s32, d32, e, compress ? 1 : 0);
  k_dinv<<<cdiv_i(n, B), B, 0, stream>>>(dinv, n);

  // ---- Layer 1 ----
  k_g1<<<cdiv_i(n, B), B, 0, stream>>>(x, W1, dinv, g, n);
  if (compress)
    k_scatter<int><<<cdiv_i(e, B), B, 0, stream>>>(s32, d32, g, out, e);
  else
    k_scatter<long long><<<cdiv_i(e, B), B, 0, stream>>>(ei, ei + e, g, out, e);
  k_finalize1<<<cdiv_i(n, B), B, 0, stream>>>(dinv, g, (float4*)out, b1, h, n);

  // ---- Layer 2 ----
  {
    int numTiles = (n + 15) / 16;
    int wavesPerBlock = B / 32;
    k_linear2_wmma<<<cdiv_i(numTiles, wavesPerBlock), B, 0, stream>>>(
        (const float*)h, W2, (float*)g, n);
  }
  if (compress)
    k_scatter<int><<<cdiv_i(e, B), B, 0, stream>>>(s32, d32, g, out, e);
  else
    k_scatter<long long><<<cdiv_i(e, B), B, 0, stream>>>(ei, ei + e, g, out, e);
  k_finalize2<<<cdiv_i(n, B), B, 0, stream>>>(dinv, g, (float4*)out, b2, n);
}